// CosineCSA_72198400246223
// MI455X (gfx1250) — compile-verified
//
#include <hip/hip_runtime.h>

// Row-wise cosine-contrastive loss, N=16384, D=2048, fp32.
// Memory-bound: 256 MB @ 23.3 TB/s ~ 11 us. One wave per row, coalesced
// float4 loads, 3 FMA accumulators per lane, cross-lane reduction via
// V_WMMA_F32_16X16X4_F32 (ones-vector trick) + one shfl_xor(16).

#define LOSS_MARGIN 0.4f
#define LOSS_EPS    1e-8f

typedef __attribute__((ext_vector_type(2))) float v2f;
typedef __attribute__((ext_vector_type(8))) float v8f;

// Full 32-lane sum of p using one f32 WMMA.
// A tile (16x4 f32): lane L, VGPR0 -> (M=L%16, K = L<16 ? 0 : 2); VGPR1 = 0.
// B = ones => D[m,n] = p[m] + p[m+16].
// D tile (16x16 f32): lanes 0-15 hold rows 0-7 in VGPR0-7, lanes 16-31 rows 8-15.
// So sum of the 8 D components per lane gives rows 0-7 (low half) / 8-15 (high
// half); one xor-16 shuffle combines the halves -> every lane has the total.
__device__ __forceinline__ float wave_allsum_wmma(float p) {
  v2f a; a.x = p;    a.y = 0.0f;
  v2f b; b.x = 1.0f; b.y = 1.0f;
  v8f c = {};
  c = __builtin_amdgcn_wmma_f32_16x16x4_f32(
      /*neg_a=*/false, a, /*neg_b=*/false, b,
      /*c_mod=*/(short)0, c, /*reuse_a=*/false, /*reuse_b=*/false);
  float s = c[0] + c[1] + c[2] + c[3] + c[4] + c[5] + c[6] + c[7];
  s += __shfl_xor(s, 16, 32);
  return s;
}

__global__ void __launch_bounds__(256)
cosine_rows_kernel(const float* __restrict__ x,
                   const float* __restrict__ y,
                   const int*   __restrict__ ceq,
                   float*       __restrict__ block_sums,
                   int N, int D) {
  const int wave = threadIdx.x >> 5;   // 8 waves per block
  const int lane = threadIdx.x & 31;

  int row = blockIdx.x * 8 + wave;
  const int valid = (row < N) ? 1 : 0;
  if (!valid) row = N - 1;             // clamp: keep EXEC full for WMMA; zeroed below

  const float* __restrict__ xr = x + (size_t)row * (size_t)D;
  const float* __restrict__ yr = y + (size_t)row * (size_t)D;

  __builtin_prefetch(xr + lane * 4, 0, 3);   // global_prefetch_b8
  __builtin_prefetch(yr + lane * 4, 0, 3);

  float ip = 0.0f, xx = 0.0f, yy = 0.0f;
  // D must be a multiple of 128 (D=2048 here): lane*4 + 128k covers the row,
  // 32 lanes x float4 = 512 contiguous bytes per wave per load.
  #pragma unroll 4
  for (int c = lane * 4; c < D; c += 128) {
    const float4 xv = *(const float4*)(xr + c);
    const float4 yv = *(const float4*)(yr + c);
    ip = fmaf(xv.x, yv.x, ip); ip = fmaf(xv.y, yv.y, ip);
    ip = fmaf(xv.z, yv.z, ip); ip = fmaf(xv.w, yv.w, ip);
    xx = fmaf(xv.x, xv.x, xx); xx = fmaf(xv.y, xv.y, xx);
    xx = fmaf(xv.z, xv.z, xx); xx = fmaf(xv.w, xv.w, xx);
    yy = fmaf(yv.x, yv.x, yy); yy = fmaf(yv.y, yv.y, yy);
    yy = fmaf(yv.z, yv.z, yy); yy = fmaf(yv.w, yv.w, yy);
  }

  // Cross-lane reductions on the matrix pipe (EXEC is all-1s here).
  const float sip = wave_allsum_wmma(ip);
  const float sxx = wave_allsum_wmma(xx);
  const float syy = wave_allsum_wmma(yy);

  // Per-row loss (all lanes compute the same value; lane 0 publishes).
  float w = sqrtf(sxx) * sqrtf(syy);
  w = fmaxf(w, LOSS_EPS);
  const float cosine = sip / w;
  const float dist = 1.0f - cosine;
  const float ce = (float)ceq[row];
  const float hinge = fmaxf(LOSS_MARGIN - dist, 0.0f);
  float loss = ce * dist * dist + (1.0f - ce) * hinge * hinge;
  loss *= (float)valid;

  __shared__ float wsum[8];
  if (lane == 0) wsum[wave] = loss;
  __syncthreads();
  if (threadIdx.x == 0) {
    float s = 0.0f;
    #pragma unroll
    for (int i = 0; i < 8; ++i) s += wsum[i];
    block_sums[blockIdx.x] = s;   // every block writes its slot: no stale ws data
  }
}

__global__ void __launch_bounds__(256)
finalize_kernel(const float* __restrict__ block_sums,
                float* __restrict__ out, int n_blocks, float inv_n) {
  __shared__ float sm[256];
  float s = 0.0f;
  for (int i = threadIdx.x; i < n_blocks; i += 256) s += block_sums[i];
  sm[threadIdx.x] = s;
  __syncthreads();
  for (int off = 128; off > 0; off >>= 1) {
    if (threadIdx.x < off) sm[threadIdx.x] += sm[threadIdx.x + off];
    __syncthreads();
  }
  if (threadIdx.x == 0) out[0] = sm[0] * inv_n;   // mean over N
}

extern "C" void kernel_launch(void* const* d_in, const int* in_sizes, int n_in,
                              void* d_out, int out_size, void* d_ws, size_t ws_size,
                              hipStream_t stream) {
  const float* x   = (const float*)d_in[0];
  const float* y   = (const float*)d_in[1];
  const int*   ceq = (const int*)d_in[2];

  const int N = in_sizes[2];            // class_eq has N elements
  const int D = in_sizes[0] / N;        // x is N*D

  float* block_sums = (float*)d_ws;     // n_blocks floats of scratch
  const int n_blocks = (N + 7) / 8;     // 8 waves (rows) per 256-thread block

  cosine_rows_kernel<<<n_blocks, 256, 0, stream>>>(x, y, ceq, block_sums, N, D);
  finalize_kernel<<<1, 256, 0, stream>>>(block_sums, (float*)d_out, n_blocks,
                                         1.0f / (float)N);
}